// FreqDecomp_32573031973593
// MI455X (gfx1250) — compile-verified
//
#include <hip/hip_runtime.h>
#include <math.h>

namespace {

constexpr int L      = 4096;         // FFT length (last axis)
constexpr int ROWS   = 16;           // rows per tile == WMMA M
constexpr int NROWS  = 64 * 128;     // 8192 independent rows
constexpr int NTILES = NROWS / ROWS; // 512
constexpr int CHUNK  = 512;          // columns staged per pass-1 WG
constexpr int NCHUNK = L / CHUNK;    // 8
constexpr int TPB    = 256;          // 8 waves (wave32)
constexpr float LAMBD = 0.01f;

typedef __attribute__((ext_vector_type(2))) float v2f;
typedef __attribute__((ext_vector_type(8))) float v8f;

__device__ __forceinline__ float softshrink(float v) {
  float m = fabsf(v) - LAMBD;
  return m > 0.0f ? copysignf(m, v) : 0.0f;
}

// Low 32 bits of a generic pointer to LDS == LDS byte address (ISA 10.2).
__device__ __forceinline__ unsigned lds_addr(const void* p) {
  return (unsigned)(unsigned long long)p;
}

// ---------------------------------------------------------------------------
// Kernel 0: basis tables cos(2*pi*n/L), sin(2*pi*n/L) into workspace.
// ---------------------------------------------------------------------------
__global__ __launch_bounds__(256) void basis_init(float* __restrict__ cosT,
                                                  float* __restrict__ sinT) {
  int n = blockIdx.x * blockDim.x + threadIdx.x;
  if (n < L) {
    double ang = 6.283185307179586476925286766559 * (double)n / (double)L;
    cosT[n] = (float)cos(ang);
    sinT[n] = (float)sin(ang);
  }
}

// ---------------------------------------------------------------------------
// Pass 1: stage x chunk -> LDS via async loads, async-store the x copy from
// LDS, and reduce per-row partial sums (s0, sum*cos, sum*sin) -> workspace.
// Grid: (NCHUNK, NTILES), 256 threads.
// ---------------------------------------------------------------------------
__global__ __launch_bounds__(TPB) void pass1_reduce_copy(
    const float* __restrict__ x, float* __restrict__ out_x,
    const float* __restrict__ cosT, const float* __restrict__ sinT,
    float4* __restrict__ part /* [NROWS][NCHUNK] */) {
  __shared__ float sh_x[ROWS * CHUNK];        // 32 KB staging
  __shared__ float red0[TPB], redc[TPB], reds[TPB];

  const int t       = threadIdx.x;
  const int c       = blockIdx.x;             // column chunk
  const int tile    = blockIdx.y;             // row tile
  const int row0    = tile * ROWS;
  const size_t gofs = (size_t)row0 * L + (size_t)c * CHUNK;

  // ---- async load: global -> LDS (b128 per lane) ----
  constexpr int RNDS = (ROWS * CHUNK) / (TPB * 4); // 8
#pragma unroll
  for (int i = 0; i < RNDS; ++i) {
    int e4 = (i * TPB + t) * 4;                // float index in chunk
    int rr = e4 >> 9;                          // / CHUNK
    int cc = e4 & (CHUNK - 1);
    unsigned lo = lds_addr(&sh_x[e4]);
    unsigned long long ga =
        (unsigned long long)(x + gofs + (size_t)rr * L + cc);
    asm volatile("global_load_async_to_lds_b128 %0, %1, off"
                 :: "v"(lo), "v"(ga) : "memory");
  }
  asm volatile("s_wait_asynccnt 0" ::: "memory");  // this wave's bytes landed

  // ---- async store: LDS -> out_x (the x-copy never touches VGPRs) ----
#pragma unroll
  for (int i = 0; i < RNDS; ++i) {
    int e4 = (i * TPB + t) * 4;
    int rr = e4 >> 9;
    int cc = e4 & (CHUNK - 1);
    unsigned lo = lds_addr(&sh_x[e4]);
    unsigned long long ga =
        (unsigned long long)(out_x + gofs + (size_t)rr * L + cc);
    asm volatile("global_store_async_from_lds_b128 %0, %1, off"
                 :: "v"(ga), "v"(lo) : "memory");
  }
  __syncthreads();  // every wave passed its own asynccnt wait -> LDS valid

  // ---- per-thread partial reduction: 16 threads/row, 32 elems each ----
  const int r = t >> 4;   // row in tile
  const int j = t & 15;   // sub-lane within row
  float s0 = 0.f, sc = 0.f, ss = 0.f;
#pragma unroll 8
  for (int kk = 0; kk < CHUNK / 16; ++kk) {
    int col = kk * 16 + j;
    float v = sh_x[r * CHUNK + col];
    int g   = c * CHUNK + col;
    s0 += v;
    sc  = fmaf(v, cosT[g], sc);
    ss  = fmaf(v, sinT[g], ss);
  }
  red0[t] = s0; redc[t] = sc; reds[t] = ss;
  __syncthreads();

  if (t < ROWS) {
    float a0 = 0.f, ac = 0.f, as = 0.f;
#pragma unroll
    for (int q = 0; q < 16; ++q) {
      a0 += red0[t * 16 + q];
      ac += redc[t * 16 + q];
      as += reds[t * 16 + q];
    }
    part[(size_t)(row0 + t) * NCHUNK + c] = make_float4(a0, ac, as, 0.0f);
  }
  // async stores drained by implicit wait-idle at s_endpgm
}

// ---------------------------------------------------------------------------
// Pass 2: finalize coefficients (softshrink) and synthesize the trend with
// V_WMMA_F32_16X16X4_F32:  C(16x16) = A(16x4 coeffs) x B(4x16 basis).
// Grid: NTILES, 256 threads. Each wave owns 32 *consecutive* 16-col chunks
// (sequential 64B store streams per row); trip count is compile-time constant
// so EXEC stays all-1s around the WMMA (no exec-masked loop).
// ---------------------------------------------------------------------------
__global__ __launch_bounds__(TPB) void pass2_trend_wmma(
    const float4* __restrict__ part, const float* __restrict__ cosT,
    const float* __restrict__ sinT, float* __restrict__ out_trend) {
  __shared__ float4 coef[ROWS];

  const int t    = threadIdx.x;
  const int tile = blockIdx.x;
  const int row0 = tile * ROWS;
  const size_t gbase = (size_t)row0 * L;

  if (t < ROWS) {
    float s0 = 0.f, sc = 0.f, ss = 0.f;
#pragma unroll
    for (int c = 0; c < NCHUNK; ++c) {
      float4 p = part[(size_t)(row0 + t) * NCHUNK + c];
      s0 += p.x; sc += p.y; ss += p.z;
    }
    const float inv = 1.0f / (float)L;
    // trend[n] = a + b*cos(theta_n) + c*sin(theta_n)
    coef[t] = make_float4(softshrink(s0) * inv,
                          2.0f * softshrink(sc) * inv,
                          2.0f * softshrink(ss) * inv, 0.0f);
  }
  __syncthreads();

  const int lane = t & 31;
  const int wave = t >> 5;
  const int nn   = lane & 15;        // N index (cols) for B/C; M row for A
  const bool hi  = lane >= 16;

  // A fragment (16x4 f32): VGPR0 = K0 (lanes 0-15) / K2 (lanes 16-31),
  //                        VGPR1 = K1 / K3.  Rows = lane&15.
  float4 cf = coef[nn];
  v2f afrag;
  afrag.x = hi ? cf.z : cf.x;        // K2 = c  | K0 = a
  afrag.y = hi ? 0.0f : cf.y;        // K3 = 0  | K1 = b

  const int mb = hi ? 8 : 0;         // C layout: lanes>=16 hold rows 8..15

  constexpr int WAVES = TPB / 32;            // 8
  constexpr int ITERS = (L / 16) / WAVES;    // 32 column chunks per wave
  const int cb0 = wave * ITERS;

  // software-pipelined basis loads (uniform clamped prefetch index)
  float ct = cosT[cb0 * 16 + nn];
  float st = sinT[cb0 * 16 + nn];

  for (int it = 0; it < ITERS; ++it) {
    int n0 = (cb0 + it) * 16;

    int itn = (it + 1 < ITERS) ? (it + 1) : it;   // uniform, no divergence
    float ctn = cosT[(cb0 + itn) * 16 + nn];
    float stn = sinT[(cb0 + itn) * 16 + nn];

    // B fragment (4x16 f32), mirroring A's K split:
    // VGPR0 = K0 row (1.0) / K2 row (sin); VGPR1 = K1 row (cos) / K3 row (0).
    v2f bfrag;
    bfrag.x = hi ? st   : 1.0f;
    bfrag.y = hi ? 0.0f : ct;

    v8f acc = {};
    acc = __builtin_amdgcn_wmma_f32_16x16x4_f32(
        /*neg_a=*/false, afrag, /*neg_b=*/false, bfrag,
        /*c_mod=*/(short)0, acc, /*reuse_a=*/false, /*reuse_b=*/false);

#pragma unroll
    for (int v = 0; v < 8; ++v)
      out_trend[gbase + (size_t)(mb + v) * L + n0 + nn] = acc[v];

    ct = ctn; st = stn;
  }
}

} // namespace

// ---------------------------------------------------------------------------
extern "C" void kernel_launch(void* const* d_in, const int* in_sizes, int n_in,
                              void* d_out, int out_size, void* d_ws,
                              size_t ws_size, hipStream_t stream) {
  (void)in_sizes; (void)n_in; (void)out_size; (void)ws_size;

  const float* x = (const float*)d_in[0];
  float* out_x     = (float*)d_out;                       // first tuple elem
  float* out_trend = out_x + (size_t)NROWS * L;           // second tuple elem

  // Workspace layout: cosT[L] | sinT[L] | part[NROWS*NCHUNK] float4 (~1.06 MB)
  float*  cosT = (float*)d_ws;
  float*  sinT = cosT + L;
  float4* part = (float4*)(sinT + L);

  basis_init<<<(L + 255) / 256, 256, 0, stream>>>(cosT, sinT);

  dim3 g1(NCHUNK, NTILES);
  pass1_reduce_copy<<<g1, TPB, 0, stream>>>(x, out_x, cosT, sinT, part);

  pass2_trend_wmma<<<NTILES, TPB, 0, stream>>>(part, cosT, sinT, out_trend);
}